// MyModel_61933428410864
// MI455X (gfx1250) — compile-verified
//
#include <hip/hip_runtime.h>
#include <hip/hip_bf16.h>

typedef __attribute__((ext_vector_type(2))) float v2f;
typedef __attribute__((ext_vector_type(8))) float v8f;

#define N_ROWS 8192
#define N_COLS 4096
#define ROWS_PER_CHUNK 256   // rows handled per wave (64 wmma steps of K=4)
#define COL_GROUPS 32        // 4096 / (8 waves * 16 cols)
#define ROW_CHUNKS (N_ROWS / ROWS_PER_CHUNK)  // 32

// ---------------------------------------------------------------------------
// Kernel 1: zero the column-sum accumulator (d_ws) every call.
// ---------------------------------------------------------------------------
__global__ void zero_kernel(float* __restrict__ s, int n) {
    int i = blockIdx.x * blockDim.x + threadIdx.x;
    if (i < n) s[i] = 0.0f;
}

// ---------------------------------------------------------------------------
// Kernel 2: column sums of relu(x) via V_WMMA_F32_16X16X4_F32.
//   s = 1^T * relu(X):  A = 16x4 ones, B = 4x16 tile of relu(x), C accumulates.
//   B layout (32-bit, 4x16 K x N): VGPR0 = {K=0 lanes 0-15, K=2 lanes 16-31},
//                                  VGPR1 = {K=1, K=3}; N = lane % 16.
//   D layout: every lane's c[0] = s_partial[col0 + lane%16] (rows of D are
//   identical because A is all ones).
// ---------------------------------------------------------------------------
__global__ __launch_bounds__(256) void colsum_wmma_kernel(
        const float* __restrict__ x, float* __restrict__ s) {
    const int lane = threadIdx.x & 31;
    const int wave = threadIdx.x >> 5;

    const int colGroup = blockIdx.x & (COL_GROUPS - 1);   // 0..31
    const int rowChunk = blockIdx.x >> 5;                 // 0..31

    const int col0  = colGroup * 128 + wave * 16;         // 16-col stripe per wave
    const int row0  = rowChunk * ROWS_PER_CHUNK;
    const int myCol = col0 + (lane & 15);
    const int rowOff = (lane >> 4) << 1;                  // 0 (lanes 0-15) or 2

    const float* p = x + (size_t)(row0 + rowOff) * N_COLS + myCol;

    v2f a;
    a[0] = 1.0f;
    a[1] = 1.0f;

    v8f c = {0.0f, 0.0f, 0.0f, 0.0f, 0.0f, 0.0f, 0.0f, 0.0f};

    #pragma unroll 4
    for (int k = 0; k < ROWS_PER_CHUNK / 4; ++k) {
        float t0 = p[0];            // row r+rowOff
        float t1 = p[N_COLS];       // row r+rowOff+1
        v2f b;
        b[0] = fmaxf(t0, 0.0f);     // relu
        b[1] = fmaxf(t1, 0.0f);
        // D = A*B + C ; 8 args: (neg_a, A, neg_b, B, c_mod, C, reuse_a, reuse_b)
        c = __builtin_amdgcn_wmma_f32_16x16x4_f32(
                false, a, false, b, (short)0, c, false, false);
        p += 4 * N_COLS;
    }

    // lanes 16-31 hold duplicate columns; commit only lanes 0-15.
    if (lane < 16) {
        atomicAdd(&s[myCol], c[0]);
    }
}

// ---------------------------------------------------------------------------
// Kernel 3: out[0] = dot(s, s), single block LDS reduction.
// ---------------------------------------------------------------------------
__global__ __launch_bounds__(256) void dot_kernel(
        const float* __restrict__ s, float* __restrict__ out, int n) {
    __shared__ float red[256];
    float acc = 0.0f;
    for (int i = threadIdx.x; i < n; i += 256) {
        float v = s[i];
        acc += v * v;
    }
    red[threadIdx.x] = acc;
    __syncthreads();
    for (int off = 128; off > 0; off >>= 1) {
        if ((int)threadIdx.x < off) red[threadIdx.x] += red[threadIdx.x + off];
        __syncthreads();
    }
    if (threadIdx.x == 0) out[0] = red[0];
}

// ---------------------------------------------------------------------------
extern "C" void kernel_launch(void* const* d_in, const int* in_sizes, int n_in,
                              void* d_out, int out_size, void* d_ws, size_t ws_size,
                              hipStream_t stream) {
    const float* x = (const float*)d_in[0];
    float* out = (float*)d_out;
    float* s = (float*)d_ws;   // 4096 floats of scratch

    // 1) zero partial-sum buffer (must happen every call)
    zero_kernel<<<(N_COLS + 255) / 256, 256, 0, stream>>>(s, N_COLS);

    // 2) WMMA column-sum of relu(x)
    colsum_wmma_kernel<<<COL_GROUPS * ROW_CHUNKS, 256, 0, stream>>>(x, s);

    // 3) dot(s, s) -> scalar
    dot_kernel<<<1, 256, 0, stream>>>(s, out, N_COLS);
}